// SlingLayer_16776142258490
// MI455X (gfx1250) — compile-verified
//
#include <hip/hip_runtime.h>
#include <hip/hip_bf16.h>

typedef __attribute__((ext_vector_type(2))) float v2f;
typedef __attribute__((ext_vector_type(8))) float v8f;

#define N_PTS  1024
#define IN_F   64
#define H_DIM  32
#define NRBF   50
#define KPAD   52          // NRBF padded to multiple of 4
#define EPS_   1e-8f
#define PI_F   3.14159265358979323846f
#define TI     16          // i-tile per block
#define TJ     8           // j-tile per block
#define PAIRS  128         // TI*TJ pairs per block

// ---------------- kernel 0: seed output with b2 * N*N ----------------
__global__ void init_out_kernel(const float* __restrict__ b2, float* __restrict__ out) {
    if (threadIdx.x == 0 && blockIdx.x == 0) {
        out[0] = b2[0] * (float)(N_PTS) * (float)(N_PTS);
    }
}

// ---------------- kernel 1: hp = h @ W_fc  (1024x64 @ 64x32) ----------------
__global__ void hp_kernel(const float* __restrict__ hin, const float* __restrict__ Wfc,
                          float* __restrict__ hp) {
    int t = blockIdx.x * blockDim.x + threadIdx.x;   // 0..32767
    int i = t >> 5;
    int hh = t & 31;
    float acc = 0.f;
#pragma unroll 8
    for (int k = 0; k < IN_F; ++k)
        acc = fmaf(hin[i * IN_F + k], Wfc[k * H_DIM + hh], acc);
    hp[t] = acc;
}

// ---------------- kernel 2: main pairwise WMMA kernel ----------------
__global__ __launch_bounds__(128)
void pair_energy_kernel(const float* __restrict__ x,
                        const float* __restrict__ hp,
                        const float* __restrict__ W_edge,
                        const float* __restrict__ means,
                        const float* __restrict__ betas,
                        const float* __restrict__ W1,
                        const float* __restrict__ b1,
                        const float* __restrict__ W2,
                        float* __restrict__ out) {
    __shared__ float s_xi[TI * 3];
    __shared__ float s_xj[TJ * 3];
    __shared__ float s_hpi[TI * H_DIM];
    __shared__ float s_hpj[TJ * H_DIM];
    __shared__ float s_We[KPAD * H_DIM];    // W_edge zero-padded rows 50..51
    __shared__ float s_W1[H_DIM * H_DIM];
    __shared__ float s_means[KPAD];
    __shared__ float s_betas[KPAD];
    __shared__ float s_dn2[PAIRS];
    __shared__ float s_cn[PAIRS * 34];      // cnorm, stride 34 (even, conflict-light)

    const int t  = threadIdx.x;             // 0..127 (4 waves, wave32)
    const int I0 = blockIdx.y * TI;
    const int J0 = blockIdx.x * TJ;

    // ---- prologue: stage tiles & weights into LDS ----
    if (t < TI * 3) s_xi[t] = x[(I0 + t / 3) * 3 + (t % 3)];
    if (t >= 64 && t < 64 + TJ * 3) {
        int u = t - 64;
        s_xj[u] = x[(J0 + u / 3) * 3 + (u % 3)];
    }
#pragma unroll
    for (int idx = t; idx < TI * H_DIM; idx += 128)
        s_hpi[idx] = hp[(I0 + (idx >> 5)) * H_DIM + (idx & 31)];
#pragma unroll
    for (int idx = t; idx < TJ * H_DIM; idx += 128)
        s_hpj[idx] = hp[(J0 + (idx >> 5)) * H_DIM + (idx & 31)];
    for (int idx = t; idx < KPAD * H_DIM; idx += 128)
        s_We[idx] = ((idx >> 5) < NRBF) ? W_edge[idx] : 0.f;
    for (int idx = t; idx < H_DIM * H_DIM; idx += 128)
        s_W1[idx] = W1[idx];
    if (t < KPAD) {
        s_means[t] = (t < NRBF) ? means[t] : 0.f;
        s_betas[t] = (t < NRBF) ? betas[t] : 0.f;
    }
    __syncthreads();

    // per-pair ||delta_n||^2 = s / (s + EPS)
    {
        int il = t >> 3, jl = t & 7;
        float dx = s_xj[jl * 3 + 0] - s_xi[il * 3 + 0];
        float dy = s_xj[jl * 3 + 1] - s_xi[il * 3 + 1];
        float dz = s_xj[jl * 3 + 2] - s_xi[il * 3 + 2];
        float s = dx * dx + dy * dy + dz * dz;
        s_dn2[t] = s / (s + EPS_);
    }
    __syncthreads();

    const int ln    = t & 15;         // lane % 16
    const int khalf = (t >> 4) & 1;   // which K-half of the wave
    const int wv    = t >> 5;         // wave id (0..3)

    const float b1n[2] = { b1[ln], b1[16 + ln] };
    const float w2n[2] = { W2[ln], W2[16 + ln] };

    float total = 0.f;

#pragma unroll
    for (int m2 = 0; m2 < 2; ++m2) {
        const int m     = wv * 2 + m2;     // M-tile id (0..7)
        const int mbase = m * 16;          // first pair of this tile

        // ---- build A fragments (rbf) in ISA A-layout, fp32, in registers ----
        const int P  = mbase + ln;         // this lane's pair (M = lane%16)
        const int il = P >> 3, jl = P & 7;
        float dx = s_xj[jl * 3 + 0] - s_xi[il * 3 + 0];
        float dy = s_xj[jl * 3 + 1] - s_xi[il * 3 + 1];
        float dz = s_xj[jl * 3 + 2] - s_xi[il * 3 + 2];
        float s    = dx * dx + dy * dy + dz * dz;
        float dist = s + EPS_;                                   // relu(s)+EPS, s>=0
        float cut  = 0.5f * (__cosf(PI_F * (0.4f * dist + 1.f)) + 1.f);
        float expd = __expf(-dist);

        v2f a[13];
#pragma unroll
        for (int tk = 0; tk < 13; ++tk) {
            int k0 = 4 * tk + 2 * khalf;                         // lane<16: 4t,4t+1 ; lane>=16: 4t+2,4t+3
            float d0 = expd - s_means[k0];
            float d1 = expd - s_means[k0 + 1];
            float r0 = cut * __expf(-s_betas[k0] * d0 * d0);
            float r1 = cut * __expf(-s_betas[k0 + 1] * d1 * d1);
            a[tk].x = (k0     < NRBF) ? r0 : 0.f;
            a[tk].y = (k0 + 1 < NRBF) ? r1 : 0.f;
        }

        // ---- GEMM1: he = rbf @ W_edge  (M=16 pairs, N=32, K=52) + elementwise ----
#pragma unroll
        for (int c = 0; c < 2; ++c) {
            v8f acc = {0.f, 0.f, 0.f, 0.f, 0.f, 0.f, 0.f, 0.f};
#pragma unroll
            for (int tk = 0; tk < 13; ++tk) {
                int kk = 4 * tk + 2 * khalf;
                v2f b;
                b.x = s_We[kk * H_DIM + 16 * c + ln];
                b.y = s_We[(kk + 1) * H_DIM + 16 * c + ln];
                acc = __builtin_amdgcn_wmma_f32_16x16x4_f32(
                    false, a[tk], false, b, (short)0, acc, false, false);
            }
            // D element (vgpr r, lane) -> pair mbase + r + 8*khalf, col 16c+ln
#pragma unroll
            for (int r = 0; r < 8; ++r) {
                int q  = mbase + r + 8 * khalf;
                int hh = 16 * c + ln;
                float ese = s_hpi[(q >> 3) * H_DIM + hh] + s_hpj[(q & 7) * H_DIM + hh];
                float cv  = ese * acc[r];
                s_cn[q * 34 + hh] = cv * cv * s_dn2[q];
            }
        }

        __syncthreads();   // cross-lane visibility of s_cn (uniform across all waves)

        // ---- GEMM2: act = cnorm @ W1 (K=32), then sigmoid . W2 partial sum ----
#pragma unroll
        for (int c2 = 0; c2 < 2; ++c2) {
            v8f acc2 = {0.f, 0.f, 0.f, 0.f, 0.f, 0.f, 0.f, 0.f};
#pragma unroll
            for (int tk = 0; tk < 8; ++tk) {
                int kk = 4 * tk + 2 * khalf;
                v2f a2, bw;
                a2.x = s_cn[(mbase + ln) * 34 + kk];
                a2.y = s_cn[(mbase + ln) * 34 + kk + 1];
                bw.x = s_W1[kk * H_DIM + 16 * c2 + ln];
                bw.y = s_W1[(kk + 1) * H_DIM + 16 * c2 + ln];
                acc2 = __builtin_amdgcn_wmma_f32_16x16x4_f32(
                    false, a2, false, bw, (short)0, acc2, false, false);
            }
#pragma unroll
            for (int r = 0; r < 8; ++r) {
                float act = acc2[r] + b1n[c2];
                total += w2n[c2] / (1.f + __expf(-act));   // sigmoid(act) * W2[n]
            }
        }
    }

    // ---- wave reduction + one atomic per wave ----
#pragma unroll
    for (int off = 16; off > 0; off >>= 1)
        total += __shfl_down(total, off, 32);
    if ((t & 31) == 0)
        atomicAdd(out, total);
}

extern "C" void kernel_launch(void* const* d_in, const int* in_sizes, int n_in,
                              void* d_out, int out_size, void* d_ws, size_t ws_size,
                              hipStream_t stream) {
    const float* h_in   = (const float*)d_in[0];
    const float* x      = (const float*)d_in[1];
    const float* W_fc   = (const float*)d_in[2];
    const float* W_edge = (const float*)d_in[3];
    const float* means  = (const float*)d_in[4];
    const float* betas  = (const float*)d_in[5];
    const float* W1     = (const float*)d_in[6];
    const float* b1     = (const float*)d_in[7];
    const float* W2     = (const float*)d_in[8];
    const float* b2     = (const float*)d_in[9];
    float* out = (float*)d_out;
    float* hp  = (float*)d_ws;   // 1024*32 floats

    init_out_kernel<<<1, 64, 0, stream>>>(b2, out);
    hp_kernel<<<(N_PTS * H_DIM) / 256, 256, 0, stream>>>(h_in, W_fc, hp);

    dim3 grid(N_PTS / TJ, N_PTS / TI);   // 128 x 64 blocks
    pair_energy_kernel<<<grid, 128, 0, stream>>>(x, hp, W_edge, means, betas,
                                                 W1, b1, W2, out);
}